// HackedTopKTopPSampler_80221399155252
// MI455X (gfx1250) — compile-verified
//
#include <hip/hip_runtime.h>
#include <hip/hip_bf16.h>
#include <math.h>

#define TPB  1024
#define NBIN 2048
#define CAP  4096
#define HCAP 1024   // |elements above bin b1| = suffix[b1+1] < k <= 1024, always fits

#if defined(__has_builtin)
#if __has_builtin(__builtin_amdgcn_global_load_async_to_lds_b128) && \
    __has_builtin(__builtin_amdgcn_s_wait_asynccnt)
#define USE_ASYNC 1
#endif
#endif
#ifndef USE_ASYNC
#define USE_ASYNC 0
#endif

#if USE_ASYNC
// Builtin signature (from hipcc diagnostic): first param is
// 'int __attribute__((vector_size(16))) __device__ *' i.e. v4i in AS(1);
// LDS destination is the AS(3) counterpart.
typedef int v4i_vs __attribute__((vector_size(16)));
typedef v4i_vs __attribute__((address_space(1))) gas_v4i;
typedef v4i_vs __attribute__((address_space(3))) las_v4i;
#endif

// ---- monotonic float <-> ordered-uint mapping (total order incl. negatives) ----
__device__ __forceinline__ unsigned ord32(float x) {
  unsigned u = __float_as_uint(x);
  return (u & 0x80000000u) ? ~u : (u | 0x80000000u);
}
__device__ __forceinline__ float unord32(unsigned u) {
  unsigned b = (u & 0x80000000u) ? (u & 0x7FFFFFFFu) : ~u;
  return __uint_as_float(b);
}
__device__ __forceinline__ unsigned hash32(unsigned x) {
  x ^= x >> 16; x *= 0x7feb352dU; x ^= x >> 15; x *= 0x846ca68bU; x ^= x >> 16;
  return x;
}
__device__ __forceinline__ unsigned long long wfix(float w) {
  return (unsigned long long)(w * 4294967296.0f);  // fixed-point 2^32, w in (0,1]
}

// inclusive suffix-sum of a[0..n-1] (result left in a); b is scratch
template <typename T>
__device__ void suffix_scan(T* a, T* b, int n) {
  T* src = a; T* dst = b;
  for (int d = 1; d < n; d <<= 1) {
    for (int i = threadIdx.x; i < n; i += TPB) {
      T v = src[i];
      if (i + d < n) v = v + src[i + d];
      dst[i] = v;
    }
    __syncthreads();
    T* t = src; src = dst; dst = t;
  }
  if (src != a) {
    for (int i = threadIdx.x; i < n; i += TPB) a[i] = src[i];
    __syncthreads();
  }
}

// find unique bin b with s[b] >= K && s[b+1] < K (suffix sums non-increasing)
template <typename T>
__device__ void find_cross(const T* s, int n, T K, int* bin, T* above) {
  for (int i = threadIdx.x; i < n; i += TPB) {
    T cur = s[i];
    T nx  = (i + 1 < n) ? s[i + 1] : (T)0;
    if (cur >= K && nx < K) { *bin = i; *above = nx; }
  }
  __syncthreads();
}

// Stream the row through a double-buffered async-to-LDS pipeline (CDNA5 path),
// calling fn(value, float4_index) for each owned element. Falls back to direct
// global_load_b128 + global_prefetch_b8 if async builtins are unavailable.
template <typename F>
__device__ void stream_row(const float4* __restrict__ x4, int nv4,
                           float4 (*stage)[TPB], F fn) {
#if USE_ASYNC
  const int tid    = threadIdx.x;
  const int ntiles = (nv4 + TPB - 1) / TPB;
  if (tid < nv4)
    __builtin_amdgcn_global_load_async_to_lds_b128(
        (gas_v4i*)(x4 + tid), (las_v4i*)&stage[0][tid], 0, 0);
  int buf = 0;
  for (int t = 0; t < ntiles; ++t) {
    const int gn = (t + 1) * TPB + tid;
    if (t + 1 < ntiles) {
      if (gn < nv4)
        __builtin_amdgcn_global_load_async_to_lds_b128(
            (gas_v4i*)(x4 + gn), (las_v4i*)&stage[buf ^ 1][tid], 0, 0);
      __builtin_amdgcn_s_wait_asynccnt(1);  // tile t landed; t+1 still in flight
    } else {
      __builtin_amdgcn_s_wait_asynccnt(0);  // drain
    }
    __syncthreads();  // publish all waves' staged data
    const int g = t * TPB + tid;
    if (g < nv4) fn(stage[buf][tid], g);
    __syncthreads();  // all reads of this buffer done before it is refilled
    buf ^= 1;
  }
#else
  for (int i = threadIdx.x; i < nv4; i += TPB) {
    __builtin_prefetch((const void*)(x4 + i + 8 * TPB), 0, 0);
    float4 v = x4[i];
    fn(v, i);
  }
#endif
}

__global__ __launch_bounds__(TPB)
void topk_topp_sample(const float* __restrict__ logits,
                      const int*   __restrict__ kvec,
                      const float* __restrict__ pvec,
                      float* __restrict__ tok_out,
                      float* __restrict__ msk_out,
                      int V) {
  const int row = blockIdx.x;
  const int tid = threadIdx.x;
  const int nv4 = V >> 2;
  const float4* __restrict__ x4 = (const float4*)(logits + (size_t)row * (size_t)V);
  float4* __restrict__ o4 = (float4*)(msk_out + (size_t)row * (size_t)V);

#if USE_ASYNC
  __shared__ float4 stage[2][TPB];             // 32 KB async staging
#else
  float4 (*stage)[TPB] = nullptr;
#endif
  __shared__ unsigned long long h64[2][NBIN];  // 32 KB; aliased as 4x u32[NBIN]
  __shared__ unsigned long long t64[NBIN];     // 16 KB scan scratch
  __shared__ float              candB[CAP];    // 16 KB: bin-b1 members
  __shared__ float              candH[HCAP];   //  4 KB: members of bins > b1
  __shared__ unsigned           s_cnt, s_hcnt;
  __shared__ int                s_bin;
  __shared__ unsigned           s_aboveU;
  __shared__ unsigned long long s_aboveW;
  __shared__ unsigned           s_umax;
  __shared__ unsigned long long s_best;

  unsigned (*h32)[NBIN] = (unsigned(*)[NBIN])h64;
  unsigned* t32 = (unsigned*)t64;

  const unsigned K1 = (unsigned)kvec[row];
  const float    p  = pvec[row];

  for (int i = tid; i < 2 * NBIN; i += TPB) ((unsigned long long*)h64)[i] = 0ull;
  if (tid == 0) { s_umax = 0u; s_best = 0ull; s_bin = -1; }
  __syncthreads();

  // ---- Pass A (HBM): count hist over top-11 ordered bits + row max ----
  {
    const int cp = (tid >> 5) & 3;  // privatized copies vs hot-bin contention
    unsigned lmax = 0u;
    stream_row(x4, nv4, stage, [&](const float4& v, int) {
      float c[4] = {v.x, v.y, v.z, v.w};
#pragma unroll
      for (int j = 0; j < 4; ++j) {
        unsigned u = ord32(c[j]);
        if (u > lmax) lmax = u;
        atomicAdd(&h32[cp][u >> 21], 1u);
      }
    });
    atomicMax(&s_umax, lmax);
    __syncthreads();
    for (int i = tid; i < NBIN; i += TPB)
      h32[0][i] += h32[1][i] + h32[2][i] + h32[3][i];
    __syncthreads();
  }
  suffix_scan<unsigned>(h32[0], t32, NBIN);
  find_cross<unsigned>(h32[0], NBIN, K1, &s_bin, &s_aboveU);
  const int      b1   = s_bin;
  const unsigned K2   = K1 - s_aboveU;   // s_aboveU = |bins > b1| < K1 <= 1024
  const unsigned umax = s_umax;
  __syncthreads();

  // ---- Pass B (L2): refine 11 bits in bin b1 + gather BOTH candidate lists ----
  for (int i = tid; i < NBIN; i += TPB) h32[0][i] = 0u;
  if (tid == 0) { s_bin = -1; s_cnt = 0u; s_hcnt = 0u; }
  __syncthreads();
  stream_row(x4, nv4, stage, [&](const float4& v, int) {
    float c[4] = {v.x, v.y, v.z, v.w};
#pragma unroll
    for (int j = 0; j < 4; ++j) {
      unsigned u = ord32(c[j]);
      int bin = (int)(u >> 21);
      if (bin == b1) {
        atomicAdd(&h32[0][(u >> 10) & 0x7FFu], 1u);
        unsigned ix = atomicAdd(&s_cnt, 1u);
        if (ix < CAP) candB[ix] = c[j];
      } else if (bin > b1) {
        unsigned ix = atomicAdd(&s_hcnt, 1u);  // guaranteed < 1024
        if (ix < HCAP) candH[ix] = c[j];
      }
    }
  });
  __syncthreads();
  const unsigned bcnt = s_cnt;
  const unsigned hcnt = s_hcnt;
  suffix_scan<unsigned>(h32[0], t32, NBIN);
  find_cross<unsigned>(h32[0], NBIN, K2, &s_bin, &s_aboveU);
  const int      b2 = s_bin;
  const unsigned K3 = K2 - s_aboveU;
  __syncthreads();

  // ---- Top-k level 3 (10 bits): LDS candidate list, or global fallback ----
  const unsigned pref21 = ((unsigned)b1 << 11) | (unsigned)b2;
  for (int i = tid; i < 1024; i += TPB) h32[0][i] = 0u;
  if (tid == 0) s_bin = -1;
  __syncthreads();
  if (bcnt <= CAP) {
    for (int i = tid; i < (int)bcnt; i += TPB) {
      unsigned u = ord32(candB[i]);
      if ((u >> 10) == pref21) atomicAdd(&h32[0][u & 0x3FFu], 1u);
    }
  } else {
    stream_row(x4, nv4, stage, [&](const float4& v, int) {
      float c[4] = {v.x, v.y, v.z, v.w};
#pragma unroll
      for (int j = 0; j < 4; ++j) {
        unsigned u = ord32(c[j]);
        if ((u >> 10) == pref21) atomicAdd(&h32[0][u & 0x3FFu], 1u);
      }
    });
  }
  __syncthreads();
  suffix_scan<unsigned>(h32[0], t32, 1024);
  find_cross<unsigned>(h32[0], 1024, K3, &s_bin, &s_aboveU);
  const unsigned ukth = (pref21 << 10) | (unsigned)s_bin;  // exact k-th largest
  __syncthreads();

  // Unified iterator over the kept set (u >= ukth). LDS-resident lists in the
  // typical case (zero global traffic); block-uniform global fallback otherwise.
  auto for_each_kept = [&](auto fn) {
    if (bcnt <= CAP) {
      for (int i = tid; i < (int)hcnt; i += TPB) fn(candH[i]);
      for (int i = tid; i < (int)bcnt; i += TPB)
        if (ord32(candB[i]) >= ukth) fn(candB[i]);
    } else {
      stream_row(x4, nv4, stage, [&](const float4& v, int) {
        float c[4] = {v.x, v.y, v.z, v.w};
#pragma unroll
        for (int j = 0; j < 4; ++j)
          if (ord32(c[j]) >= ukth) fn(c[j]);
      });
    }
  };

  // ---- Nucleus selection: 3-level weighted radix (u64 fixed point) ----
  const float m = unord32(umax);

  // level 1: 2048 bins over top-11 ordered bits
  for (int i = tid; i < 2 * NBIN; i += TPB) ((unsigned long long*)h64)[i] = 0ull;
  if (tid == 0) s_bin = -1;
  __syncthreads();
  {
    const int cp = (tid >> 5) & 1;
    for_each_kept([&](float xv) {
      atomicAdd(&h64[cp][ord32(xv) >> 21], wfix(__expf(xv - m)));
    });
  }
  __syncthreads();
  for (int i = tid; i < NBIN; i += TPB) h64[0][i] += h64[1][i];
  __syncthreads();
  suffix_scan<unsigned long long>(h64[0], t64, NBIN);
  const unsigned long long Sfix = h64[0][0];
  unsigned long long pS = (unsigned long long)((double)p * (double)Sfix);
  if (pS < 1ull) pS = 1ull;
  find_cross<unsigned long long>(h64[0], NBIN, pS, &s_bin, &s_aboveW);
  const int                c1     = s_bin;
  const unsigned long long above1 = s_aboveW;
  __syncthreads();

  unsigned upth;
  if (c1 < 0) {
    upth = umax;  // degenerate: keep only the max logit
  } else {
    // level 2: next 11 bits within bin c1
    const unsigned long long target2 = pS - above1;  // >= 1
    for (int i = tid; i < NBIN; i += TPB) h64[0][i] = 0ull;
    if (tid == 0) s_bin = -1;
    __syncthreads();
    for_each_kept([&](float xv) {
      unsigned u = ord32(xv);
      if ((int)(u >> 21) == c1)
        atomicAdd(&h64[0][(u >> 10) & 0x7FFu], wfix(__expf(xv - m)));
    });
    __syncthreads();
    suffix_scan<unsigned long long>(h64[0], t64, NBIN);
    find_cross<unsigned long long>(h64[0], NBIN, target2, &s_bin, &s_aboveW);
    const int                c2     = (s_bin < 0) ? 0 : s_bin;
    const unsigned long long above2 = (s_bin < 0) ? 0ull : s_aboveW;
    __syncthreads();

    // level 3: final 10 bits -> exact 32-bit nucleus boundary
    unsigned long long target3 = target2 - above2;
    if (target3 < 1ull) target3 = 1ull;
    const unsigned cpref = ((unsigned)c1 << 11) | (unsigned)c2;
    for (int i = tid; i < 1024; i += TPB) h64[0][i] = 0ull;
    if (tid == 0) s_bin = -1;
    __syncthreads();
    for_each_kept([&](float xv) {
      unsigned u = ord32(xv);
      if ((u >> 10) == cpref)
        atomicAdd(&h64[0][u & 0x3FFu], wfix(__expf(xv - m)));
    });
    __syncthreads();
    suffix_scan<unsigned long long>(h64[0], t64, 1024);
    find_cross<unsigned long long>(h64[0], 1024, target3, &s_bin, &s_aboveW);
    const int c3 = (s_bin < 0) ? 0 : s_bin;
    upth = (cpref << 10) | (unsigned)c3;
    __syncthreads();
  }
  const unsigned ukeep = (ukth > upth) ? ukth : upth;

  // ---- Pass D (L2): write masked logits + fused Gumbel-style argmax ----
  stream_row(x4, nv4, stage, [&](const float4& v, int gi) {
    float c[4] = {v.x, v.y, v.z, v.w};
    float r[4];
#pragma unroll
    for (int j = 0; j < 4; ++j) {
      const int col = 4 * gi + j;
      unsigned u = ord32(c[j]);
      bool keep = (u >= ukeep) || (u == umax);  // max logit always kept
      r[j] = keep ? c[j] : -INFINITY;
      if (keep) {
        unsigned h   = hash32(((unsigned)(row * V + col)) * 2654435761u ^ 0xA511E9B3u);
        float    u01 = (float)((h >> 8) + 1u) * (1.0f / 16777216.0f);  // (0,1]
        float    sc  = c[j] + __logf(u01);  // x - Exp(1) noise
        unsigned long long key =
            ((unsigned long long)ord32(sc) << 32) | (unsigned)(V - 1 - col);
        atomicMax(&s_best, key);
      }
    }
    o4[gi] = make_float4(r[0], r[1], r[2], r[3]);
  });
  __syncthreads();
  if (tid == 0) {
    unsigned col = (unsigned)(V - 1) - (unsigned)(s_best & 0xFFFFFFFFull);
    tok_out[row] = (float)col;
  }
}

extern "C" void kernel_launch(void* const* d_in, const int* in_sizes, int n_in,
                              void* d_out, int out_size, void* d_ws, size_t ws_size,
                              hipStream_t stream) {
  const float* logits = (const float*)d_in[0];
  const int*   k      = (const int*)d_in[1];
  const float* p      = (const float*)d_in[2];
  const int B = in_sizes[1];
  const int V = in_sizes[0] / B;
  float* tok = (float*)d_out;      // tokens first (B), then masked logits (B*V)
  float* msk = tok + B;
  hipLaunchKernelGGL(topk_topp_sample, dim3(B), dim3(TPB), 0, stream,
                     logits, k, p, tok, msk, V);
}